// HeteroGATConvGNN_70729521430712
// MI455X (gfx1250) — compile-verified
//
#include <hip/hip_runtime.h>
#include <hip/hip_bf16.h>
#include <math.h>

// ---------------------------------------------------------------------------
// HeteroGATv2 (2 layers, user<->movie) for MI455X (gfx1250, wave32, WMMA)
//  H = 64, E = 1,000,000 per direction, self-loops = min(n_src,n_dst) = 20,000
// ---------------------------------------------------------------------------

typedef __attribute__((ext_vector_type(2))) float v2f;
typedef __attribute__((ext_vector_type(8))) float v8f;

#define N_USER  100000
#define N_MOVIE 20000
#define HDIM    64
#define NEDGE   1000000
#define NLOOP   20000     // min(N_USER, N_MOVIE)
#define NLAYER  2

#define NEG_SLOPE 0.2f
#define SELU_SCALE 1.0507009873554805f
#define SELU_ALPHA 1.6732632423543772f

// ---- order-preserving float <-> uint mapping for atomic max -----------------
__device__ __forceinline__ unsigned f2ord(float f) {
  unsigned u = __float_as_uint(f);
  return (u & 0x80000000u) ? ~u : (u | 0x80000000u);
}
__device__ __forceinline__ float ord2f(unsigned u) {
  return __uint_as_float((u & 0x80000000u) ? (u ^ 0x80000000u) : ~u);
}

// ---------------------------------------------------------------------------
// GEMM: Y[N x 64] = X[N x 64] @ W[64 x 64] + b[64], via V_WMMA_F32_16X16X4_F32
// One wave computes one 16x16 output tile; 16 WMMA steps cover K = 64.
// A layout (16x4 f32): lanes 0-15 -> M=lane, v0=K0,v1=K1; lanes 16-31 -> K2,K3
// B layout (4x16 f32): lanes 0-15 -> N=lane, v0=K0,v1=K1; lanes 16-31 -> K2,K3
// C/D layout: VGPR j: lanes 0-15 -> M=j, lanes 16-31 -> M=j+8; N = lane&15
// ---------------------------------------------------------------------------
__global__ __launch_bounds__(256) void gemm64_bias_wmma(
    const float* __restrict__ X, const float* __restrict__ W,
    const float* __restrict__ b, float* __restrict__ Y, int nRowTiles)
{
  const int wave = (int)((blockIdx.x * blockDim.x + threadIdx.x) >> 5);
  const int lane = (int)(threadIdx.x & 31);
  const int nTiles = nRowTiles * 4;                 // 4 column tiles of 16
  if (wave >= nTiles) return;                       // wave-uniform guard

  const int tr   = wave >> 2;                       // row tile
  const int tc   = wave & 3;                        // col tile
  const int half = lane >> 4;                       // 0: K{0,1}, 1: K{2,3}
  const int m    = lane & 15;                       // A row / B-C-D column
  const int col  = tc * 16 + m;

  const float* __restrict__ xrow = X + (size_t)(tr * 16 + m) * HDIM + 2 * half;
  const float* __restrict__ wcol = W + (size_t)(2 * half) * HDIM + col;

  v8f c = {};
#pragma unroll
  for (int k = 0; k < HDIM; k += 4) {
    v2f a;
    a.x = xrow[k + 0];
    a.y = xrow[k + 1];
    v2f bv;
    bv.x = wcol[(size_t)k * HDIM];
    bv.y = wcol[(size_t)k * HDIM + HDIM];
    c = __builtin_amdgcn_wmma_f32_16x16x4_f32(
        /*neg_a=*/false, a, /*neg_b=*/false, bv,
        /*c_mod=*/(short)0, c, /*reuse_a=*/false, /*reuse_b=*/false);
  }

  const float bias = b[col];
#pragma unroll
  for (int j = 0; j < 8; ++j) {
    const int row = tr * 16 + j + 8 * half;
    Y[(size_t)row * HDIM + col] = c[j] + bias;
  }
}

// ---------------------------------------------------------------------------
// Per-dst init: maxb = ord(-inf), sumb = 0, agg = 0
// ---------------------------------------------------------------------------
__global__ __launch_bounds__(256) void init_seg_kernel(
    unsigned* __restrict__ maxb, float* __restrict__ sumb,
    float* __restrict__ agg, int n_dst)
{
  const int i = (int)(blockIdx.x * blockDim.x + threadIdx.x);
  const int tot = n_dst * HDIM;
  if (i < n_dst) {
    maxb[i] = 0x007FFFFFu;   // f2ord(-inf)
    sumb[i] = 0.0f;
  }
  if (i < tot) agg[i] = 0.0f;
}

// ---------------------------------------------------------------------------
// Pass A: score[e] = att . leaky_relu(xl[src] + xr[dst]); atomic max per dst.
// One wave32 per edge; lane covers 2 of the 64 features.
// ---------------------------------------------------------------------------
__global__ __launch_bounds__(256) void edge_score_kernel(
    const float* __restrict__ xl, const float* __restrict__ xr,
    const float* __restrict__ att,
    const int* __restrict__ srcA, const int* __restrict__ dstA,
    int nEdge, int nLoop, float* __restrict__ score, unsigned* __restrict__ maxb)
{
  const int e    = (int)((blockIdx.x * blockDim.x + threadIdx.x) >> 5);
  const int lane = (int)(threadIdx.x & 31);
  const int tot  = nEdge + nLoop;
  if (e >= tot) return;

  const int s = (e < nEdge) ? srcA[e] : (e - nEdge);   // self loop: src=dst=i
  const int d = (e < nEdge) ? dstA[e] : (e - nEdge);

  const v2f xlv = *(const v2f*)(xl + (size_t)s * HDIM + lane * 2);
  const v2f xrv = *(const v2f*)(xr + (size_t)d * HDIM + lane * 2);

  float h0 = xlv.x + xrv.x; h0 = (h0 > 0.0f) ? h0 : NEG_SLOPE * h0;
  float h1 = xlv.y + xrv.y; h1 = (h1 > 0.0f) ? h1 : NEG_SLOPE * h1;
  float sc = h0 * att[lane * 2] + h1 * att[lane * 2 + 1];

#pragma unroll
  for (int off = 16; off > 0; off >>= 1)
    sc += __shfl_down(sc, off, 32);                    // wave32 reduce

  if (lane == 0) {
    score[e] = sc;
    atomicMax(maxb + d, f2ord(sc));
  }
}

// ---------------------------------------------------------------------------
// Pass B: score[e] <- exp(score[e] - max[dst]); atomic sum per dst.
// ---------------------------------------------------------------------------
__global__ __launch_bounds__(256) void edge_exp_kernel(
    float* __restrict__ score, const int* __restrict__ dstA,
    int nEdge, int nLoop,
    const unsigned* __restrict__ maxb, float* __restrict__ sumb)
{
  const int e   = (int)(blockIdx.x * blockDim.x + threadIdx.x);
  const int tot = nEdge + nLoop;
  if (e >= tot) return;
  const int d = (e < nEdge) ? dstA[e] : (e - nEdge);
  const float m  = ord2f(maxb[d]);
  const float ex = __expf(score[e] - m);
  score[e] = ex;
  atomicAdd(sumb + d, ex);
}

// ---------------------------------------------------------------------------
// Pass C: agg[dst] += xl[src] * (ex / (sum[dst] + 1e-16)); wave32 per edge.
// ---------------------------------------------------------------------------
__global__ __launch_bounds__(256) void edge_agg_kernel(
    const float* __restrict__ xl, const float* __restrict__ score,
    const float* __restrict__ sumb,
    const int* __restrict__ srcA, const int* __restrict__ dstA,
    int nEdge, int nLoop, float* __restrict__ agg)
{
  const int e    = (int)((blockIdx.x * blockDim.x + threadIdx.x) >> 5);
  const int lane = (int)(threadIdx.x & 31);
  const int tot  = nEdge + nLoop;
  if (e >= tot) return;

  const int s = (e < nEdge) ? srcA[e] : (e - nEdge);
  const int d = (e < nEdge) ? dstA[e] : (e - nEdge);

  const float alpha = score[e] / (sumb[d] + 1e-16f);
  const v2f xlv = *(const v2f*)(xl + (size_t)s * HDIM + lane * 2);
  float* p = agg + (size_t)d * HDIM + lane * 2;
  atomicAdd(p + 0, xlv.x * alpha);
  atomicAdd(p + 1, xlv.y * alpha);
}

// ---------------------------------------------------------------------------
// out = selu(agg + bias)
// ---------------------------------------------------------------------------
__global__ __launch_bounds__(256) void bias_selu_kernel(
    const float* __restrict__ agg, const float* __restrict__ bias,
    float* __restrict__ out, int n_dst)
{
  const int i = (int)(blockIdx.x * blockDim.x + threadIdx.x);
  if (i >= n_dst * HDIM) return;
  const float v = agg[i] + bias[i & (HDIM - 1)];
  out[i] = (v > 0.0f) ? SELU_SCALE * v
                      : SELU_SCALE * SELU_ALPHA * (__expf(v) - 1.0f);
}

// ---------------------------------------------------------------------------
// Host-side: one GATv2 conv (single edge type)
// ---------------------------------------------------------------------------
static void run_gatv2(hipStream_t stream,
                      const float* x_src, int n_src,
                      const float* x_dst, int n_dst,
                      const int* srcA, const int* dstA,
                      const float* Wl, const float* bl,
                      const float* Wr, const float* br,
                      const float* att, const float* bias,
                      float* xl, float* xr, float* score,
                      unsigned* maxb, float* sumb,
                      float* agg, float* out)
{
  const int tot = NEDGE + NLOOP;

  // projections (WMMA f32 GEMMs)
  {
    int waves = (n_src / 16) * 4;
    gemm64_bias_wmma<<<(waves + 7) / 8, 256, 0, stream>>>(x_src, Wl, bl, xl, n_src / 16);
  }
  {
    int waves = (n_dst / 16) * 4;
    gemm64_bias_wmma<<<(waves + 7) / 8, 256, 0, stream>>>(x_dst, Wr, br, xr, n_dst / 16);
  }

  // segment state init
  {
    int n = n_dst * HDIM;
    init_seg_kernel<<<(n + 255) / 256, 256, 0, stream>>>(maxb, sumb, agg, n_dst);
  }

  // scores + segment max  (wave per edge)
  edge_score_kernel<<<(tot + 7) / 8, 256, 0, stream>>>(
      xl, xr, att, srcA, dstA, NEDGE, NLOOP, score, maxb);

  // exp + segment sum  (thread per edge)
  edge_exp_kernel<<<(tot + 255) / 256, 256, 0, stream>>>(
      score, dstA, NEDGE, NLOOP, maxb, sumb);

  // weighted aggregation  (wave per edge)
  edge_agg_kernel<<<(tot + 7) / 8, 256, 0, stream>>>(
      xl, score, sumb, srcA, dstA, NEDGE, NLOOP, agg);

  // bias + SELU
  {
    int n = n_dst * HDIM;
    bias_selu_kernel<<<(n + 255) / 256, 256, 0, stream>>>(agg, bias, out, n_dst);
  }
}

extern "C" void kernel_launch(void* const* d_in, const int* in_sizes, int n_in,
                              void* d_out, int out_size, void* d_ws, size_t ws_size,
                              hipStream_t stream) {
  (void)in_sizes; (void)n_in; (void)out_size; (void)ws_size;

  const float* in_xu  = (const float*)d_in[0];          // [N_USER, 64]
  const float* in_xm  = (const float*)d_in[1];          // [N_MOVIE, 64]
  const int*   e_u2m  = (const int*)d_in[2];            // [2, E] flat
  const int*   e_m2u  = (const int*)d_in[3];            // [2, E] flat
  const float* Wl     = (const float*)d_in[4];          // [L,2,64,64]
  const float* bl     = (const float*)d_in[5];          // [L,2,64]
  const float* Wr     = (const float*)d_in[6];
  const float* br     = (const float*)d_in[7];
  const float* att    = (const float*)d_in[8];          // [L,2,64]
  const float* bias   = (const float*)d_in[9];          // [L,2,64]

  float* out_user  = (float*)d_out;                          // [N_USER, 64]
  float* out_movie = (float*)d_out + (size_t)N_USER * HDIM;  // [N_MOVIE, 64]

  // ---- workspace carving (floats) ----
  float* ws = (float*)d_ws;
  size_t o = 0;
  float* cur_user  = ws + o; o += (size_t)N_USER  * HDIM;
  float* cur_movie = ws + o; o += (size_t)N_MOVIE * HDIM;
  float* xl        = ws + o; o += (size_t)N_USER  * HDIM;   // max(n_src)
  float* xr        = ws + o; o += (size_t)N_USER  * HDIM;   // max(n_dst)
  float* agg_m     = ws + o; o += (size_t)N_MOVIE * HDIM;
  float* agg_u     = ws + o; o += (size_t)N_USER  * HDIM;
  float* score     = ws + o; o += (size_t)(NEDGE + NLOOP);
  unsigned* maxb   = (unsigned*)(ws + o); o += (size_t)N_USER;
  float* sumb      = ws + o; o += (size_t)N_USER;

  for (int l = 0; l < NLAYER; ++l) {
    const float* xu = (l == 0) ? in_xu : cur_user;
    const float* xm = (l == 0) ? in_xm : cur_movie;
    float* tgt_movie = (l == NLAYER - 1) ? out_movie : cur_movie;
    float* tgt_user  = (l == NLAYER - 1) ? out_user  : cur_user;

    // edge type 0: user -> movie
    {
      const int idx = l * 2 + 0;
      run_gatv2(stream,
                xu, N_USER, xm, N_MOVIE,
                e_u2m, e_u2m + NEDGE,
                Wl + (size_t)idx * HDIM * HDIM, bl + (size_t)idx * HDIM,
                Wr + (size_t)idx * HDIM * HDIM, br + (size_t)idx * HDIM,
                att + (size_t)idx * HDIM, bias + (size_t)idx * HDIM,
                xl, xr, score, maxb, sumb, agg_m, tgt_movie);
    }
    // edge type 1: movie -> user (reads OLD xm — tgt_movie != xm source here)
    {
      const int idx = l * 2 + 1;
      run_gatv2(stream,
                xm, N_MOVIE, xu, N_USER,
                e_m2u, e_m2u + NEDGE,
                Wl + (size_t)idx * HDIM * HDIM, bl + (size_t)idx * HDIM,
                Wr + (size_t)idx * HDIM * HDIM, br + (size_t)idx * HDIM,
                att + (size_t)idx * HDIM, bias + (size_t)idx * HDIM,
                xl, xr, score, maxb, sumb, agg_u, tgt_user);
    }
  }
}